// MeshFit_49185965474289
// MI455X (gfx1250) — compile-verified
//
#include <hip/hip_runtime.h>
#include <hip/hip_bf16.h>
#include <limits.h>

typedef __attribute__((ext_vector_type(2))) float v2f;
typedef __attribute__((ext_vector_type(8))) float v8f;

constexpr int CLS = 4;      // classes
constexpr int NV  = 4096;   // old vertices per class
constexpr int MV  = 4096;   // new vertices per class
constexpr int FD  = 32;     // feature dim (== wave32 width)

struct DI { float d; int i; };
__device__ __forceinline__ DI di_min(DI a, DI b) {
  bool t = (a.d < b.d) || (a.d == b.d && a.i < b.i);
  return t ? a : b;
}
__device__ __forceinline__ DI di_max(DI a, DI b) {
  bool t = (a.d < b.d) || (a.d == b.d && a.i < b.i);
  return t ? b : a;
}
__device__ __forceinline__ int imin32(int a, int b) { return a < b ? a : b; }
__device__ __forceinline__ int imax32(int a, int b) { return a > b ? a : b; }

// One wave owns a 16-row tile of new vertices of one class and scans all 4096
// class vertices (staged in LDS) in 16x16 WMMA tiles, keeping a per-lane sorted
// top-3 of (truncated-dist | column) keys, then butterfly-merges across lanes.
// Keys compare as SIGNED ints: fp-cancellation-negative distances (true d ~ 0)
// sort first automatically, so no clamp is needed in the hot loop.
__global__ __launch_bounds__(128)
void meshfit_topk_wmma(const float* __restrict__ feat,    // [C*N][D]
                       const float* __restrict__ verts,   // [C][N][3]
                       const float* __restrict__ nverts,  // [C][M][3]
                       float* __restrict__ out)           // [C*M][D]
{
  __shared__ float4 svtx[NV];  // 64 KB: (x, y, z, |v|^2)

  const int tid  = threadIdx.x;
  const int lane = tid & 31;
  const int wave = tid >> 5;
  const int blk  = blockIdx.x;
  const int c    = blk >> 6;    // 64 blocks per class
  const int mb   = blk & 63;

  // ---- stage this class's vertices (with precomputed norms) into LDS ----
  const float* vc = verts + (size_t)c * NV * 3;
  for (int n = tid; n < NV; n += 128) {
    float x = vc[n * 3 + 0], y = vc[n * 3 + 1], z = vc[n * 3 + 2];
    svtx[n] = make_float4(x, y, z, x * x + y * y + z * z);
  }
  __syncthreads();

  const int half = lane >> 4;                // K-half for WMMA operand layout
  const int l16  = lane & 15;
  const int m0   = mb * 64 + wave * 16;      // row-tile base within class

  // ---- A matrix (16x4): row l16 = new vertex, K = [x, y, z, 1] ----
  const float* av = nverts + ((size_t)c * MV + (m0 + l16)) * 3;
  const float ax = av[0], ay = av[1], az = av[2];
  const float na = ax * ax + ay * ay + az * az;
  v2f A;
  A[0] = half ? az : ax;       // lanes 0-15: (K0,K1)=(x,y); 16-31: (K2,K3)=(z,1)
  A[1] = half ? 1.0f : ay;

  // ---- accumulator preload: C[m][n] = |a_m|^2 (rows half*8 .. half*8+7) ----
  v8f Cn;
#pragma unroll
  for (int v = 0; v < 8; ++v)
    Cn[v] = __shfl(na, half * 8 + v, 32);

  // ---- per-lane sorted top-3 signed keys: (dist bits & ~0xFFF) | n_local ----
  int k0[8], k1[8], k2[8];
#pragma unroll
  for (int v = 0; v < 8; ++v) { k0[v] = INT_MAX; k1[v] = INT_MAX; k2[v] = INT_MAX; }

  // software-pipelined B column: prefetch next tile's vertex while crunching
  float4 b = svtx[l16];
#pragma unroll 2
  for (int t = 0; t < NV / 16; ++t) {
    const int tn = (t + 1 < NV / 16) ? (t + 1) : t;   // uniform, clamped
    const float4 bnext = svtx[tn * 16 + l16];

    const int n = t * 16 + l16;
    v2f B;                      // col l16, K = [-2x, -2y, -2z, |b|^2]
    B[0] = half ? (-2.0f * b.z) : (-2.0f * b.x);
    B[1] = half ? b.w           : (-2.0f * b.y);
    // D = A*B + C = |a|^2 - 2 a.b + |b|^2  -> 16x16 tile of squared dists
    v8f dmat = __builtin_amdgcn_wmma_f32_16x16x4_f32(
        false, A, false, B, (short)0, Cn, false, false);
#pragma unroll
    for (int v = 0; v < 8; ++v) {
      const int key = (int)((__float_as_uint(dmat[v]) & 0xFFFFF000u) | (unsigned)n);
      int lo  = imin32(key, k0[v]);
      int hi  = imax32(key, k0[v]);
      k0[v] = lo;
      int lo1 = imin32(hi, k1[v]);
      int hi1 = imax32(hi, k1[v]);
      k1[v] = lo1;
      k2[v] = imin32(hi1, k2[v]);
    }
    b = bnext;
  }

  // ---- butterfly merge of sorted triples across the 16 lanes per half ----
#pragma unroll
  for (int off = 1; off < 16; off <<= 1) {
#pragma unroll
    for (int v = 0; v < 8; ++v) {
      int b0 = __shfl_xor(k0[v], off, 32);
      int b1 = __shfl_xor(k1[v], off, 32);
      int b2 = __shfl_xor(k2[v], off, 32);
      int r0 = imin32(k0[v], b0);
      int x  = imax32(k0[v], b0);
      int y  = imin32(k1[v], b1);
      int z  = imax32(k1[v], b1);
      int w  = imin32(k2[v], b2);
      int r1 = imin32(x, y);
      int r2 = imin32(imax32(x, y), imin32(z, w));
      k0[v] = r0; k1[v] = r1; k2[v] = r2;
    }
  }

  // ---- each lane finalizes row (half*8 + lane%8) ----
  const int v = lane & 7;
  DI a0{ __uint_as_float((unsigned)k0[v] & 0xFFFFF000u), (int)((unsigned)k0[v] & 0xFFFu) + c * NV };
  DI a1{ __uint_as_float((unsigned)k1[v] & 0xFFFFF000u), (int)((unsigned)k1[v] & 0xFFFu) + c * NV };
  DI a2{ __uint_as_float((unsigned)k2[v] & 0xFFFFF000u), (int)((unsigned)k2[v] & 0xFFFu) + c * NV };
  // off-block entries of the full dist matrix are exactly 1.0; the three
  // smallest-index ones are the only possible top-3 contenders.
  const int ob = (c == 0) ? NV : 0;
  DI ob0{1.0f, ob}, ob1{1.0f, ob + 1}, ob2{1.0f, ob + 2};
  DI r0 = di_min(a0, ob0);
  DI x  = di_max(a0, ob0);
  DI y  = di_min(a1, ob1);
  DI z  = di_max(a1, ob1);
  DI w  = di_min(a2, ob2);
  DI r1 = di_min(x, y);
  DI r2 = di_min(di_max(x, y), di_min(z, w));

  // softmax(-d) over the 3 neighbors
  float e0 = expf(-r0.d), e1 = expf(-r1.d), e2 = expf(-r2.d);
  float inv = 1.0f / (e0 + e1 + e2);
  float w0 = e0 * inv, w1 = e1 * inv, w2 = e2 * inv;
  int   j0 = r0.i,     j1 = r1.i,     j2 = r2.i;

  // ---- gather + weighted sum: lane == feature channel (D == 32) ----
  for (int r = 0; r < 16; ++r) {
    const int owner = (r < 8) ? r : (r + 8);   // lanes 0-7 own rows 0-7; 16-23 rows 8-15
    float ww0 = __shfl(w0, owner, 32);
    float ww1 = __shfl(w1, owner, 32);
    float ww2 = __shfl(w2, owner, 32);
    int   jj0 = __shfl(j0, owner, 32);
    int   jj1 = __shfl(j1, owner, 32);
    int   jj2 = __shfl(j2, owner, 32);
    float o = ww0 * feat[(size_t)jj0 * FD + lane]
            + ww1 * feat[(size_t)jj1 * FD + lane]
            + ww2 * feat[(size_t)jj2 * FD + lane];
    out[((size_t)(c * MV + m0 + r)) * FD + lane] = o;
  }
}

extern "C" void kernel_launch(void* const* d_in, const int* in_sizes, int n_in,
                              void* d_out, int out_size, void* d_ws, size_t ws_size,
                              hipStream_t stream) {
  (void)in_sizes; (void)n_in; (void)out_size; (void)d_ws; (void)ws_size;
  const float* feat   = (const float*)d_in[0];  // [1, C*N, D] float32
  const float* verts  = (const float*)d_in[1];  // [C, N, 3]  float32
  const float* nverts = (const float*)d_in[2];  // [C, M, 3]  float32
  float* out = (float*)d_out;                   // [1, C*M, D] float32

  dim3 grid(CLS * (MV / 64));  // 256 workgroups (64 per class)
  dim3 block(128);             // 4 waves x 16-row tiles = 64 rows per WG
  hipLaunchKernelGGL(meshfit_topk_wmma, grid, block, 0, stream,
                     feat, verts, nverts, out);
}